// GATv2GraphLayer_27960237097202
// MI455X (gfx1250) — compile-verified
//
#include <hip/hip_runtime.h>
#include <cmath>

typedef __attribute__((ext_vector_type(2))) float v2f;
typedef __attribute__((ext_vector_type(8))) float v8f;
typedef __attribute__((ext_vector_type(4))) unsigned int v4u;
typedef __attribute__((ext_vector_type(8))) int v8i;
typedef __attribute__((ext_vector_type(4))) int v4i;

#define HIDc 128
#define Hc 4
#define Dc 32

// ---------------------------------------------------------------------------
// TDM: DMA a contiguous 16x128 fp32 tile from global memory into LDS.
// D# descriptor per CDNA5 ISA ch.8 (2-D tile -> groups 2/3 disabled/zero).
// Issue from ONE wave; caller waits s_wait_tensorcnt then block-barriers.
// This toolchain exposes the 6-arg builtin form.
// ---------------------------------------------------------------------------
__device__ __forceinline__ void tdm_load_tile_16x128(const float* gsrc,
                                                     void* lds_dst) {
  const unsigned long long ga = (unsigned long long)(uintptr_t)gsrc;
  const unsigned int lofs = (unsigned int)(uintptr_t)lds_dst;  // LDS byte offset

  v4u g0;
  g0.x = 1u;                                      // count=1 valid user D#
  g0.y = lofs;                                    // lds_addr
  g0.z = (unsigned int)ga;                        // global_addr[31:0]
  g0.w = (unsigned int)(ga >> 32) | (2u << 30);   // global_addr[56:32] | type=2

  v8i g1;
  g1[0] = (int)(2u << 16);     // workgroup_mask=0, data_size=2 (4B)
  g1[1] = (int)(128u << 16);   // tensor_dim0 = 128 (low 16 bits at [63:48])
  g1[2] = (int)(16u << 16);    // tensor_dim1 = 16  (low 16 bits at [95:80])
  g1[3] = (int)(128u << 16);   // tile_dim0 = 128   ([127:112])
  g1[4] = 16;                  // tile_dim1 = 16    ([143:128]); tile_dim2 = 0
  g1[5] = 128;                 // tensor_dim0_stride = 128 elements
  g1[6] = 0;                   // tensor_dim1_stride = 0 (unused, 2-D)
  g1[7] = 0;

  v4i gz4 = {0, 0, 0, 0};
  v8i gz8 = {};
  __builtin_amdgcn_tensor_load_to_lds(g0, g1, gz4, gz4, gz8, 0);
}

// ---------------------------------------------------------------------------
// float atomic max via int/uint trick (buffer must be initialized to -inf)
// ---------------------------------------------------------------------------
__device__ __forceinline__ void atomicMaxFloat(float* addr, float val) {
  if (val >= 0.0f)
    atomicMax((int*)addr, __float_as_int(val));
  else
    atomicMin((unsigned int*)addr, __float_as_uint(val));
}

// ---------------------------------------------------------------------------
// init: zero agg + sums, set mx = -inf
// ---------------------------------------------------------------------------
__global__ void init_kernel(float* __restrict__ agg, float* __restrict__ mx,
                            float* __restrict__ sums, int n128, int n4) {
  int i = blockIdx.x * blockDim.x + threadIdx.x;
  if (i < n128) agg[i] = 0.0f;
  if (i < n4) { sums[i] = 0.0f; mx[i] = -__builtin_inff(); }
}

// ---------------------------------------------------------------------------
// FiLM: film = task @ W_film + b_film ; gscale = 1 + 0.5*tanh(gamma); beta
// ---------------------------------------------------------------------------
__global__ void film_kernel(const float* __restrict__ task,
                            const float* __restrict__ Wf,
                            const float* __restrict__ bf,
                            float* __restrict__ gscale,
                            float* __restrict__ beta) {
  int j = threadIdx.x;              // 0..255
  float acc = bf[j];
  for (int k = 0; k < HIDc; ++k) acc += task[k] * Wf[k * 2 * HIDc + j];
  if (j < HIDc) gscale[j] = 1.0f + 0.5f * tanhf(acc);
  else          beta[j - HIDc] = acc;
}

// ---------------------------------------------------------------------------
// Input GEMMs: x_src = X @ W_src + b_src ; x_dst = X @ W_dst + b_dst
// block = 256 threads = 8 waves; block computes 16 rows x 128 cols via WMMA.
// A tile staged into LDS by the Tensor Data Mover (wave 0 issues).
// ---------------------------------------------------------------------------
__global__ __launch_bounds__(256) void gemm_in_kernel(
    const float* __restrict__ X,
    const float* __restrict__ Ws, const float* __restrict__ bs,
    const float* __restrict__ Wd, const float* __restrict__ bd,
    float* __restrict__ xs, float* __restrict__ xd) {
  const float* W = blockIdx.y ? Wd : Ws;
  const float* b = blockIdx.y ? bd : bs;
  float* O       = blockIdx.y ? xd : xs;
  const int row0 = blockIdx.x * 16;

  __shared__ float lA[16 * HIDc];
  const int t = threadIdx.x;
  const int wave = t >> 5, lane = t & 31;

  if (wave == 0) {
    tdm_load_tile_16x128(X + (size_t)row0 * HIDc, lA);
    __builtin_amdgcn_s_wait_tensorcnt(0);
  }
  __syncthreads();

  const int col0 = wave * 16;
  const int mrow = lane & 15;   // M (A) / N (B,C)
  const int khalf = lane >> 4;  // K sub-offset

  v8f acc = {};
  for (int k0 = 0; k0 < HIDc; k0 += 4) {
    const int kk = k0 + 2 * khalf;
    v2f a, bf;
    a.x  = lA[mrow * HIDc + kk];
    a.y  = lA[mrow * HIDc + kk + 1];
    bf.x = W[(size_t)kk * HIDc + col0 + mrow];
    bf.y = W[(size_t)(kk + 1) * HIDc + col0 + mrow];
    acc = __builtin_amdgcn_wmma_f32_16x16x4_f32(false, a, false, bf,
                                                (short)0, acc, false, false);
  }

  const int col = col0 + mrow;
  const float bias = b[col];
#pragma unroll
  for (int v = 0; v < 8; ++v) {
    const int r = row0 + v + 8 * khalf;
    O[(size_t)r * HIDc + col] = acc[v] + bias;
  }
}

// ---------------------------------------------------------------------------
// Edge pass 1: logits[e][h] = dot(att[h], leaky(xs[src]+xd[dst]+eemb[type]))
// and segment-max into mx[dst][h]. One wave32 per edge; lane handles h*32+lane.
// ---------------------------------------------------------------------------
__global__ __launch_bounds__(256) void edge_logits_kernel(
    const float* __restrict__ xs, const float* __restrict__ xd,
    const float* __restrict__ eemb, const float* __restrict__ att,
    const int* __restrict__ src, const int* __restrict__ dst,
    const int* __restrict__ etype,
    float* __restrict__ logits, float* __restrict__ mx, int E_) {
  const int wave = threadIdx.x >> 5, lane = threadIdx.x & 31;
  const int e = blockIdx.x * 8 + wave;
  if (e >= E_) return;
  const int s = src[e], d = dst[e], ty = etype[e];

  float c[Hc];
#pragma unroll
  for (int h = 0; h < Hc; ++h) {
    const int idx = h * Dc + lane;
    float v = xs[(size_t)s * HIDc + idx] + xd[(size_t)d * HIDc + idx] +
              eemb[(size_t)ty * HIDc + idx];
    v = v > 0.0f ? v : 0.2f * v;           // leaky_relu(0.2)
    c[h] = v * att[idx];
  }
#pragma unroll
  for (int h = 0; h < Hc; ++h)
    for (int off = 16; off; off >>= 1)
      c[h] += __shfl_xor(c[h], off, 32);

  if (lane < Hc) {
    logits[(size_t)e * Hc + lane] = c[lane];
    atomicMaxFloat(&mx[(size_t)d * Hc + lane], c[lane]);
  }
}

// ---------------------------------------------------------------------------
// Edge pass 2: ex = exp(logit - fix(mx[dst])); segment-sum into sums[dst]
// logits buffer is overwritten in place with ex. One thread per edge.
// ---------------------------------------------------------------------------
__global__ void edge_exp_kernel(const int* __restrict__ dst,
                                float* __restrict__ logits,
                                const float* __restrict__ mx,
                                float* __restrict__ sums, int E_) {
  const int e = blockIdx.x * blockDim.x + threadIdx.x;
  if (e >= E_) return;
  const int d = dst[e];
#pragma unroll
  for (int h = 0; h < Hc; ++h) {
    float m = mx[(size_t)d * Hc + h];
    if (isinf(m)) m = 0.0f;
    const float ex = __expf(logits[(size_t)e * Hc + h] - m);
    logits[(size_t)e * Hc + h] = ex;
    atomicAdd(&sums[(size_t)d * Hc + h], ex);
  }
}

// ---------------------------------------------------------------------------
// Edge pass 3: agg[dst] += x_src[src] * attn ; wave32 per edge
// ---------------------------------------------------------------------------
__global__ __launch_bounds__(256) void edge_scatter_kernel(
    const float* __restrict__ xs, const int* __restrict__ src,
    const int* __restrict__ dst, const float* __restrict__ ex,
    const float* __restrict__ sums, float* __restrict__ agg, int E_) {
  const int wave = threadIdx.x >> 5, lane = threadIdx.x & 31;
  const int e = blockIdx.x * 8 + wave;
  if (e >= E_) return;
  const int s = src[e], d = dst[e];
#pragma unroll
  for (int h = 0; h < Hc; ++h) {
    const float a =
        ex[(size_t)e * Hc + h] / fmaxf(sums[(size_t)d * Hc + h], 1e-12f);
    const int idx = h * Dc + lane;
    atomicAdd(&agg[(size_t)d * HIDc + idx],
              xs[(size_t)s * HIDc + idx] * a);
  }
}

// ---------------------------------------------------------------------------
// Output: (agg @ W_out + b_out) * gscale + beta + node ; then row LayerNorm.
// block = 256 threads = 8 waves; 16 rows x 128 cols per block (WMMA + LDS).
// A tile staged by TDM.
// ---------------------------------------------------------------------------
__global__ __launch_bounds__(256) void out_kernel(
    const float* __restrict__ agg, const float* __restrict__ Wo,
    const float* __restrict__ bo, const float* __restrict__ node,
    const float* __restrict__ gscale, const float* __restrict__ beta,
    const float* __restrict__ nw, const float* __restrict__ nb,
    float* __restrict__ out) {
  const int row0 = blockIdx.x * 16;
  __shared__ float lA[16 * HIDc];
  __shared__ float lY[16 * HIDc];
  const int t = threadIdx.x;
  const int wave = t >> 5, lane = t & 31;

  if (wave == 0) {
    tdm_load_tile_16x128(agg + (size_t)row0 * HIDc, lA);
    __builtin_amdgcn_s_wait_tensorcnt(0);
  }
  __syncthreads();

  const int col0 = wave * 16;
  const int mrow = lane & 15, khalf = lane >> 4;

  v8f acc = {};
  for (int k0 = 0; k0 < HIDc; k0 += 4) {
    const int kk = k0 + 2 * khalf;
    v2f a, bf;
    a.x  = lA[mrow * HIDc + kk];
    a.y  = lA[mrow * HIDc + kk + 1];
    bf.x = Wo[(size_t)kk * HIDc + col0 + mrow];
    bf.y = Wo[(size_t)(kk + 1) * HIDc + col0 + mrow];
    acc = __builtin_amdgcn_wmma_f32_16x16x4_f32(false, a, false, bf,
                                                (short)0, acc, false, false);
  }

  const int col = col0 + mrow;
  const float bias = bo[col], gs = gscale[col], bt = beta[col];
#pragma unroll
  for (int v = 0; v < 8; ++v) {
    const int r = v + 8 * khalf;
    const float y = (acc[v] + bias) * gs + bt +
                    node[(size_t)(row0 + r) * HIDc + col];
    lY[r * HIDc + col] = y;
  }
  __syncthreads();

  // LayerNorm: 16 lanes per row (rows t>>4); shfl within 16-lane halves
  const int r = t >> 4;
  const int c0 = t & 15;
  float s = 0.0f, sq = 0.0f;
#pragma unroll
  for (int j = 0; j < 8; ++j) {
    const float v = lY[r * HIDc + c0 + 16 * j];
    s += v; sq += v * v;
  }
  for (int off = 8; off; off >>= 1) {
    s  += __shfl_xor(s, off, 32);
    sq += __shfl_xor(sq, off, 32);
  }
  const float mu  = s * (1.0f / HIDc);
  const float var = sq * (1.0f / HIDc) - mu * mu;
  const float inv = rsqrtf(var + 1e-5f);
#pragma unroll
  for (int j = 0; j < 8; ++j) {
    const int c = c0 + 16 * j;
    const float v = lY[r * HIDc + c];
    out[(size_t)(row0 + r) * HIDc + c] = (v - mu) * inv * nw[c] + nb[c];
  }
}

// ---------------------------------------------------------------------------
extern "C" void kernel_launch(void* const* d_in, const int* in_sizes, int n_in,
                              void* d_out, int out_size, void* d_ws,
                              size_t ws_size, hipStream_t stream) {
  const float* node  = (const float*)d_in[0];
  const int*   eidx  = (const int*)d_in[1];
  const int*   etype = (const int*)d_in[2];
  const float* task  = (const float*)d_in[3];
  const float* Wsrc  = (const float*)d_in[4];
  const float* bsrc  = (const float*)d_in[5];
  const float* Wdst  = (const float*)d_in[6];
  const float* bdst  = (const float*)d_in[7];
  const float* eemb  = (const float*)d_in[8];
  const float* att   = (const float*)d_in[9];
  const float* Wout  = (const float*)d_in[10];
  const float* bout  = (const float*)d_in[11];
  const float* nw    = (const float*)d_in[12];
  const float* nb    = (const float*)d_in[13];
  const float* Wfilm = (const float*)d_in[14];
  const float* bfilm = (const float*)d_in[15];

  const int Nn = in_sizes[0] / HIDc;   // 50000
  const int Ee = in_sizes[2];          // 500000
  const int* src = eidx;
  const int* dst = eidx + Ee;

  // workspace carve-up (floats)
  float* ws     = (float*)d_ws;
  float* xs     = ws;
  float* xd     = xs + (size_t)Nn * HIDc;
  float* agg    = xd + (size_t)Nn * HIDc;
  float* logits = agg + (size_t)Nn * HIDc;
  float* mx     = logits + (size_t)Ee * Hc;
  float* sums   = mx + (size_t)Nn * Hc;
  float* gscale = sums + (size_t)Nn * Hc;
  float* beta   = gscale + HIDc;

  const int rowTiles = (Nn + 15) / 16;           // 3125

  init_kernel<<<(Nn * HIDc + 255) / 256, 256, 0, stream>>>(
      agg, mx, sums, Nn * HIDc, Nn * Hc);

  gemm_in_kernel<<<dim3(rowTiles, 2), 256, 0, stream>>>(
      node, Wsrc, bsrc, Wdst, bdst, xs, xd);

  film_kernel<<<1, 256, 0, stream>>>(task, Wfilm, bfilm, gscale, beta);

  edge_logits_kernel<<<(Ee + 7) / 8, 256, 0, stream>>>(
      xs, xd, eemb, att, src, dst, etype, logits, mx, Ee);

  edge_exp_kernel<<<(Ee + 255) / 256, 256, 0, stream>>>(
      dst, logits, mx, sums, Ee);

  edge_scatter_kernel<<<(Ee + 7) / 8, 256, 0, stream>>>(
      xs, src, dst, logits, sums, agg, Ee);

  out_kernel<<<rowTiles, 256, 0, stream>>>(
      agg, Wout, bout, node, gscale, beta, nw, nb, (float*)d_out);
}